// Tucker_32238024524552
// MI455X (gfx1250) — compile-verified
//
#include <hip/hip_runtime.h>

typedef float v2f __attribute__((ext_vector_type(2)));
typedef float v8f __attribute__((ext_vector_type(8)));

#define RES_A 128
#define RES_B 256

// d_ws layout (float offsets). Transposed grids: gt[i*128 + (d*8+c)] = g[d][c][i]
#define OFF_G0A 0        // 128*128
#define OFF_G1A 16384
#define OFF_G2A 32768
#define OFF_G0B 49152    // 256*128
#define OFF_G1B 81920
#define OFF_G2B 114688
#define OFF_CORE_A 147456 // 16x64 zero-padded core (rows x=8..15 are 0)
#define OFF_CORE_B 148480
// total = 149504 floats = 598,016 bytes of scratch

__global__ void transpose_grid_kernel(const float* __restrict__ src,
                                      float* __restrict__ dst, int r) {
  int idx = blockIdx.x * blockDim.x + threadIdx.x;  // over 128*r elements
  if (idx < 128 * r) {
    int i = idx >> 7;      // resolution index
    int f = idx & 127;     // d*8 + c
    dst[idx] = src[f * r + i];   // coalesced write, strided (cached) read
  }
}

__global__ void pad_core_kernel(const float* __restrict__ src,
                                float* __restrict__ dst) {
  int idx = blockIdx.x * blockDim.x + threadIdx.x;  // 1024
  if (idx < 1024) dst[idx] = (idx < 512) ? src[idx] : 0.0f;
}

// One wave32 per (point, scale). 8 waves / block -> 4 points / block.
__global__ __launch_bounds__(256) void tucker_wmma_kernel(
    const float* __restrict__ pts, const float* __restrict__ ws,
    float* __restrict__ out) {
  __shared__ float stage[8][3 * 128];   // per-wave staging of ix|iy|iz (16x8 each)

  const int tid   = threadIdx.x;
  const int w     = tid >> 5;
  const int lane  = tid & 31;
  const int point = blockIdx.x * 4 + (w >> 1);
  const int scale = w & 1;

  const int   r     = scale ? RES_B : RES_A;
  const float rm1   = (float)(r - 1);
  const float* g0   = ws + (scale ? OFF_G0B : OFF_G0A);
  const float* g1   = ws + (scale ? OFF_G1B : OFF_G1A);
  const float* g2   = ws + (scale ? OFF_G2B : OFF_G2A);
  const float* coreP = ws + (scale ? OFF_CORE_B : OFF_CORE_A);  // 16x64, x-major

  float* sw = &stage[w][0];

  // ---- interpolate the 3 axes; indices are wave-uniform -> coalesced b128 column loads
#pragma unroll
  for (int ax = 0; ax < 3; ++ax) {
    const float* gt = (ax == 0) ? g0 : (ax == 1) ? g1 : g2;
    float c   = pts[point * 3 + ax];
    float pos = (c + 1.0f) * 0.5f * rm1;
    pos = fminf(fmaxf(pos, 0.0f), rm1);
    float i0f = floorf(pos);
    int   i0  = (int)i0f;
    int   i1  = i0 + 1; if (i1 > r - 1) i1 = r - 1;
    float t   = pos - i0f;
    const float4 c0 = *reinterpret_cast<const float4*>(gt + i0 * 128 + lane * 4);
    const float4 c1 = *reinterpret_cast<const float4*>(gt + i1 * 128 + lane * 4);
    float4 v;
    v.x = fmaf(t, c1.x - c0.x, c0.x);
    v.y = fmaf(t, c1.y - c0.y, c0.y);
    v.z = fmaf(t, c1.z - c0.z, c0.z);
    v.w = fmaf(t, c1.w - c0.w, c0.w);
    *reinterpret_cast<float4*>(sw + ax * 128 + lane * 4) = v;
  }
  // LDS is in-order within a wave; fence the compiler + explicit ds wait.
  __builtin_amdgcn_wave_barrier();
  asm volatile("s_wait_dscnt 0" ::: "memory");

  const int d  = lane & 15;   // A-matrix row this lane feeds; also C-column it holds
  const int hi = lane >> 4;

  const float* ixBuf = sw;          // axis 0 -> core dim x
  const float* iyBuf = sw + 128;    // axis 1 -> core dim y
  const float* izBuf = sw + 256;    // axis 2 -> core dim z

  // per-lane operands for u[d, k] = iy[d, k>>3] * iz[d, k&7], k = 4c + 2*hi + {0,1}
  float4 iy03 = *reinterpret_cast<const float4*>(iyBuf + d * 8);
  float4 iy47 = *reinterpret_cast<const float4*>(iyBuf + d * 8 + 4);
  float2 ze   = *reinterpret_cast<const float2*>(izBuf + d * 8 + 2 * hi);       // c even
  float2 zo   = *reinterpret_cast<const float2*>(izBuf + d * 8 + 4 + 2 * hi);   // c odd
  float iyv[8] = {iy03.x, iy03.y, iy03.z, iy03.w, iy47.x, iy47.y, iy47.z, iy47.w};

  // epilogue operand: ix[d_out = j+8*hi, x = lane&7] (cols 8..15 hit zeroed acc)
  float ixv[8];
#pragma unroll
  for (int j = 0; j < 8; ++j)
    ixv[j] = ixBuf[(j + 8 * hi) * 8 + (lane & 7)];

  // ---- t3[d, x] = u(16x64) @ core^T(64x16)   via 16x v_wmma_f32_16x16x4_f32
  v8f acc0 = {0.f, 0.f, 0.f, 0.f, 0.f, 0.f, 0.f, 0.f};
  v8f acc1 = {0.f, 0.f, 0.f, 0.f, 0.f, 0.f, 0.f, 0.f};
  const float* bbase = coreP + d * 64 + 2 * hi;  // B[k,n]: lane n=d, k = 4c+2*hi,+1

#pragma unroll
  for (int c = 0; c < 16; ++c) {
    float2 bp = *reinterpret_cast<const float2*>(bbase + 4 * c);
    v2f b; b.x = bp.x; b.y = bp.y;
    float z0 = (c & 1) ? zo.x : ze.x;
    float z1 = (c & 1) ? zo.y : ze.y;
    v2f a; a.x = iyv[c >> 1] * z0; a.y = iyv[c >> 1] * z1;
    if (c & 1)
      acc1 = __builtin_amdgcn_wmma_f32_16x16x4_f32(false, a, false, b, (short)0,
                                                   acc1, false, false);
    else
      acc0 = __builtin_amdgcn_wmma_f32_16x16x4_f32(false, a, false, b, (short)0,
                                                   acc0, false, false);
  }

  // ---- out[d_out] = sum_x t3[d_out, x] * ix[d_out, x]; reduce 16-lane halves
  float red[8];
#pragma unroll
  for (int j = 0; j < 8; ++j) {
    float v = (acc0[j] + acc1[j]) * ixv[j];
    v += __shfl_xor(v, 1, 32);
    v += __shfl_xor(v, 2, 32);
    v += __shfl_xor(v, 4, 32);
    v += __shfl_xor(v, 8, 32);
    red[j] = v;
  }

  if ((lane & 15) == 0) {
    float* dst = out + point * 32 + scale * 16 + 8 * hi;
    *reinterpret_cast<float4*>(dst)     = make_float4(red[0], red[1], red[2], red[3]);
    *reinterpret_cast<float4*>(dst + 4) = make_float4(red[4], red[5], red[6], red[7]);
  }
}

extern "C" void kernel_launch(void* const* d_in, const int* in_sizes, int n_in,
                              void* d_out, int out_size, void* d_ws, size_t ws_size,
                              hipStream_t stream) {
  (void)n_in; (void)out_size; (void)ws_size;
  const float* pts   = (const float*)d_in[0];
  const float* g0a   = (const float*)d_in[1];
  const float* g1a   = (const float*)d_in[2];
  const float* g2a   = (const float*)d_in[3];
  const float* corea = (const float*)d_in[4];
  const float* g0b   = (const float*)d_in[5];
  const float* g1b   = (const float*)d_in[6];
  const float* g2b   = (const float*)d_in[7];
  const float* coreb = (const float*)d_in[8];
  float* ws  = (float*)d_ws;
  float* out = (float*)d_out;

  // Prep: transpose grids to column-contiguous, zero-pad cores to 16x64.
  transpose_grid_kernel<<<64, 256, 0, stream>>>(g0a, ws + OFF_G0A, RES_A);
  transpose_grid_kernel<<<64, 256, 0, stream>>>(g1a, ws + OFF_G1A, RES_A);
  transpose_grid_kernel<<<64, 256, 0, stream>>>(g2a, ws + OFF_G2A, RES_A);
  transpose_grid_kernel<<<128, 256, 0, stream>>>(g0b, ws + OFF_G0B, RES_B);
  transpose_grid_kernel<<<128, 256, 0, stream>>>(g1b, ws + OFF_G1B, RES_B);
  transpose_grid_kernel<<<128, 256, 0, stream>>>(g2b, ws + OFF_G2B, RES_B);
  pad_core_kernel<<<4, 256, 0, stream>>>(corea, ws + OFF_CORE_A);
  pad_core_kernel<<<4, 256, 0, stream>>>(coreb, ws + OFF_CORE_B);

  int n = in_sizes[0] / 3;                 // 131072 points
  tucker_wmma_kernel<<<n / 4, 256, 0, stream>>>(pts, ws, out);
}